// ModelBaseline_27487790694641
// MI455X (gfx1250) — compile-verified
//
#include <hip/hip_runtime.h>

typedef __attribute__((ext_vector_type(2))) float v2f;
typedef __attribute__((ext_vector_type(8))) float v8f;

#define NCOD 64
#define BLOCK 256

// One block per RNA row. Stream 2048 int32 codon IDs with int4 loads,
// accumulate weighted sum (LDS table lookup) + nonzero count, then reduce
// the 256 partial sums / 256 partial counts with V_WMMA_F32_16X16X4_F32
// (A = partials, B = ones => row-sum reduction on the matrix pipe).
__global__ __launch_bounds__(BLOCK) void codon_freq_dot_kernel(
    const int* __restrict__ rna, const float* __restrict__ W,
    const float* __restrict__ bias, float* __restrict__ out, int S)
{
    __shared__ float table[NCOD + 8];   // W_ext: table[0]=0, table[v]=W[v-1]
    __shared__ float partials[2 * BLOCK]; // [0..255]=sums, [256..511]=counts

    const int tid = threadIdx.x;
    if (tid <= NCOD) table[tid] = (tid == 0) ? 0.0f : W[tid - 1];
    __syncthreads();

    const long long row = blockIdx.x;
    const int4* rp4 = (const int4*)(rna + row * (long long)S);
    const int n4 = S >> 2;

    float sum = 0.0f;
    int   cnt = 0;
    for (int i = tid; i < n4; i += BLOCK) {
        int4 v = rp4[i];                         // global_load_b128, coalesced
        sum += table[v.x] + table[v.y] + table[v.z] + table[v.w];
        cnt += (v.x != 0) + (v.y != 0) + (v.z != 0) + (v.w != 0);
    }
    partials[tid]         = sum;
    partials[BLOCK + tid] = (float)cnt;          // <= 2048, exact in f32
    __syncthreads();

    // Wave 0 (lanes 0..31, EXEC all-1s within the wave) does the WMMA reduce.
    if (tid < 32) {
        const int m     = tid & 15;              // A-matrix row for this lane
        const int kbase = (tid >> 4) << 1;       // lanes 0-15: K=0,1; 16-31: K=2,3

        v8f c = {};                              // f32 16x16 accumulator
        v2f ones; ones.x = 1.0f; ones.y = 1.0f;  // B = 4x16 all-ones

        #pragma unroll
        for (int it = 0; it < 8; ++it) {
            // rows 0-7 consume sum partials, rows 8-15 consume count partials;
            // (it, m, k) -> 32*it + (m&7)*4 + k covers 0..255 bijectively.
            int base = (m < 8) ? (32 * it + m * 4 + kbase)
                               : (BLOCK + 32 * it + (m - 8) * 4 + kbase);
            v2f a; a.x = partials[base]; a.y = partials[base + 1];
            // D[m][n] = sum_k A[m][k] * 1 + C[m][n]  (row-sum accumulate)
            c = __builtin_amdgcn_wmma_f32_16x16x4_f32(
                    false, a, false, ones, (short)0, c, false, false);
        }

        // Lane 0: VGPRs 0..7 = rows 0..7 (sum pieces).
        // Lane 16: VGPRs 0..7 = rows 8..15 (count pieces).
        float local = c[0] + c[1] + c[2] + c[3] + c[4] + c[5] + c[6] + c[7];
        float tsum = __shfl(local, 0, 32);
        float tcnt = __shfl(local, 16, 32);
        if (tid == 0)
            out[row] = tsum / tcnt + bias[0];
    }
}

extern "C" void kernel_launch(void* const* d_in, const int* in_sizes, int n_in,
                              void* d_out, int out_size, void* d_ws, size_t ws_size,
                              hipStream_t stream) {
    (void)n_in; (void)d_ws; (void)ws_size;
    const int*   rna  = (const int*)d_in[0];   // (B, S) int32 codon ids
    // d_in[1] = tissue_id: unused by the reference computation
    const float* W    = (const float*)d_in[2]; // (1, 64) weights
    const float* bias = (const float*)d_in[3]; // (1,)   bias
    float*       out  = (float*)d_out;         // (B, 1) float32

    const int B = in_sizes[1];                 // 16384 (tissue_id length)
    const int S = in_sizes[0] / B;             // 2048

    codon_freq_dot_kernel<<<B, BLOCK, 0, stream>>>(rna, W, bias, out, S);
    (void)out_size;
}